// AdjLieODEflow__816043786310
// MI455X (gfx1250) — compile-verified
//
#include <hip/hip_runtime.h>

// ---------------------------------------------------------------------------
// AdjLieODEflow on MI455X (gfx1250): B=131072 independent 8x8 Lie-ODEs, RK4 x16.
// Compute-bound (AI ~700 flop/B): all 8x8x8 fp32 matmuls run on
// V_WMMA_F32_16X16X4_F32, packing 2 batch elements per wave into 16x16 tiles
// and fusing operand-sharing matmuls into multi-block tiles (~70% MAC util).
// LDS is the layout-conversion fabric; each wave is self-contained -> the
// in-order DS pipeline means no barriers are needed, only compiler fences.
// ---------------------------------------------------------------------------

typedef float v2f __attribute__((ext_vector_type(2)));
typedef float v8f __attribute__((ext_vector_type(8)));

// prevent the compiler from reordering LDS ops (HW DS pipe is in-order per wave)
#define LDS_FENCE() asm volatile("" ::: "memory")

__device__ __forceinline__ v8f wmma4(v2f a, v2f b, v8f c) {
  // D(16x16,f32) = A(16x4,f32) x B(4x16,f32) + C
  return __builtin_amdgcn_wmma_f32_16x16x4_f32(false, a, false, b, (short)0, c,
                                               false, false);
}

// Operand layout conventions (f32, wave32):
//  A 16x4 : lane l (m=l&15) holds A[m][kb], A[m][kb+1] with kb=(l>>4)*2
//  B 4x16 : lane l (n=l&15) holds B[kb][n], B[kb+1][n]  (mirrors A; assumed)
//  C 16x16: vgpr r, lane l: row = r + 8*(l>=16), col = l&15
// K=8 products are two chained K=4 WMMAs (global k = j*4 + kb, j=0,1).

__global__ __launch_bounds__(256) void ode_kernel(
    const float* __restrict__ U0g, const float* __restrict__ epsg,
    const float* __restrict__ W0g, const float* __restrict__ W1g,
    const float* __restrict__ W2g, float* __restrict__ outU,
    float* __restrict__ outLJ) {
  __shared__ float ldsAll[8][1024];  // 4 KB per wave
  const int tid  = threadIdx.x;
  const int wv   = tid >> 5;
  const int lane = tid & 31;
  const int half = lane >> 4;
  const int l16  = lane & 15;
  const int kb   = half * 2;
  float* W = ldsAll[wv];

  const int b0 = (blockIdx.x * 8 + wv) * 2;
  const int b1 = b0 + 1;

  // ---- per-wave LDS map (float offsets) ----
  // UV0 @   0 : Ustage_b0[64] | V_b0[64]      (row-major 8x8 each)
  // UV1 @ 128 : Ustage_b1[64] | V_b1[64]
  // AP0 @ 256 : A_b0[64]      | P_b0[64]      (A = U@W2, P = V@W2)
  // AP1 @ 384 : A_b1[64]      | P_b1[64]
  // C2  @ 512 : scratch tile b0[256], b1[256]

  // ---- init: global -> LDS ----
  {
    v2f u0 = *(const v2f*)(U0g  + (size_t)b0 * 64 + lane * 2);
    v2f u1 = *(const v2f*)(U0g  + (size_t)b1 * 64 + lane * 2);
    v2f e0 = *(const v2f*)(epsg + (size_t)b0 * 64 + lane * 2);
    v2f e1 = *(const v2f*)(epsg + (size_t)b1 * 64 + lane * 2);
    *(v2f*)(W +   0 + lane * 2) = u0;
    *(v2f*)(W +  64 + lane * 2) = e0;
    *(v2f*)(W + 128 + lane * 2) = u1;
    *(v2f*)(W + 192 + lane * 2) = e1;
  }
  LDS_FENCE();

  // ---- persistent register operands ----
  // B-operand [W2 | W2] (8x16), and antisym(W0), antisym(W1) as A-operand
  // rows 0..7 (F block) / zero for rows 8..15 (dF block has no W term).
  v2f w2b[2], aw0[2], aw1[2];
#pragma unroll
  for (int j = 0; j < 2; ++j) {
    const int k0 = j * 4 + kb;
    w2b[j].x = W2g[k0 * 8 + (l16 & 7)];
    w2b[j].y = W2g[(k0 + 1) * 8 + (l16 & 7)];
    if (l16 < 8) {
      const int m = l16;
      aw0[j].x = 0.5f * (W0g[m * 8 + k0]     - W0g[k0 * 8 + m]);
      aw0[j].y = 0.5f * (W0g[m * 8 + k0 + 1] - W0g[(k0 + 1) * 8 + m]);
      aw1[j].x = 0.5f * (W1g[m * 8 + k0]     - W1g[k0 * 8 + m]);
      aw1[j].y = 0.5f * (W1g[m * 8 + k0 + 1] - W1g[(k0 + 1) * 8 + m]);
    } else {
      aw0[j].x = aw0[j].y = 0.f;
      aw1[j].x = aw1[j].y = 0.f;
    }
  }

  // Hutchinson weights in C-tile layout: lane n in [8,16): V[r][n-8] (F@V blk),
  // lane n in [16,24): V[r][n-16] (dF@U blk), else 0.
  float Vw[2][8];
  {
    int cc = -1;
    if (lane >= 8 && lane < 16) cc = lane - 8;
    else if (lane >= 16 && lane < 24) cc = lane - 16;
#pragma unroll
    for (int r = 0; r < 8; ++r) {
      Vw[0][r] = (cc >= 0) ? epsg[(size_t)b0 * 64 + r * 8 + cc] : 0.f;
      Vw[1][r] = (cc >= 0) ? epsg[(size_t)b1 * 64 + r * 8 + cc] : 0.f;
    }
  }

  // RK state: U_b rows in vgpr r, cols in lanes 0..7
  float Ur[2][8];
#pragma unroll
  for (int r = 0; r < 8; ++r) {
    Ur[0][r] = (lane < 8) ? U0g[(size_t)b0 * 64 + r * 8 + lane] : 0.f;
    Ur[1][r] = (lane < 8) ? U0g[(size_t)b1 * 64 + r * 8 + lane] : 0.f;
  }

  // ---- precomputed lane-invariant LDS addresses (float offsets) ----
  const int  g1a = l16 * 8 + ((l16 >= 8) ? 64 : 0) + kb;  // stacked [Us0;Us1] rows
  const int  pva = g1a + 64;                              // stacked [V0;V1] rows
  const bool predDiag = (lane < 8) || (lane >= 24);       // diag blocks of C
  const int  astoA = (lane < 8) ? (256 + lane) : (384 + (l16 - 8));  // + r*8
  const int  astoP = astoA + 64;
  const int  g2a[2] = {256 + l16 * 8 + kb, 384 + l16 * 8 + kb};  // [A;P] rows
  const int  g2b[2] = {  0 + l16 * 8 + kb, 128 + l16 * 8 + kb};  // [Uᵀ|Vᵀ] cols
  const int  c2st[2]   = {512 + half * 128 + l16, 768 + half * 128 + l16};
  const int  c2base[2] = {512, 768};
  const int  colp = (l16 & 7) + ((l16 >= 8) ? 64 : 0);    // [U|V] B-op col part
  const int  uvb[2] = {0, 128};

  // ---- precompute P_b = V_b @ W2 once (constant for whole ODE) ----
  {
    v8f c = {};
#pragma unroll
    for (int j = 0; j < 2; ++j) {
      v2f a = *(const v2f*)(W + pva + j * 4);
      c = wmma4(a, w2b[j], c);
    }
    if (predDiag) {
#pragma unroll
      for (int r = 0; r < 8; ++r) W[astoP + r * 8] = c[r];
    }
    LDS_FENCE();
  }

  // ---- one RHS eval for both batch elements: k_u and Hutchinson partials ----
  auto evalRHS = [&](float t, float (&kout)[2][8], float (&ss)[2]) {
    // G1: [Us0;Us1] @ [W2|W2] -> A_b0, A_b1 on the diagonal blocks
    v8f c1 = {};
#pragma unroll
    for (int j = 0; j < 2; ++j) {
      v2f a = *(const v2f*)(W + g1a + j * 4);
      c1 = wmma4(a, w2b[j], c1);
    }
    if (predDiag) {
#pragma unroll
      for (int r = 0; r < 8; ++r) W[astoA + r * 8] = c1[r];
    }
    LDS_FENCE();

#pragma unroll
    for (int b = 0; b < 2; ++b) {
      // G2: [A;P] @ [Usᵀ|Vᵀ] -> [Mu, R; Q, junk]  (transposes free via addressing)
      v8f c2 = {};
#pragma unroll
      for (int j = 0; j < 2; ++j) {
        v2f a  = *(const v2f*)(W + g2a[b] + j * 4);
        v2f bb = *(const v2f*)(W + g2b[b] + j * 4);
        c2 = wmma4(a, bb, c2);
      }
#pragma unroll
      for (int r = 0; r < 8; ++r) W[c2st[b] + r * 16] = c2[r];
      LDS_FENCE();

      // Build stacked A-operand [F;dF]:
      //   F[m][k]  = wt[m][k] + 0.5*(Mu[m][k] - Mu[k][m])      (rows 0..7)
      //   dF[i][k] = 0.5*(Q[i][k]+R[i][k] - Q[k][i] - R[k][i]) (rows 8..15)
      // Unified per lane: v = wt + 0.5*(L1 - L4) + s*(L2 - L3), s = 0.5 for dF rows.
      v2f fd[2];
      const int   C2b = c2base[b];
      const float sdf = (l16 >= 8) ? 0.5f : 0.f;
#pragma unroll
      for (int j = 0; j < 2; ++j) {
        const int k0 = j * 4 + kb;
        v2f   L1  = *(const v2f*)(W + C2b + l16 * 16 + k0);
        v2f   L2  = *(const v2f*)(W + C2b + (l16 & 7) * 16 + 8 + k0);
        float L3x = W[C2b + (8 + k0) * 16 + (l16 & 7)];
        float L3y = W[C2b + (9 + k0) * 16 + (l16 & 7)];
        float L4x = W[C2b + k0 * 16 + l16];
        float L4y = W[C2b + (k0 + 1) * 16 + l16];
        fd[j].x = fmaf(t, aw1[j].x, aw0[j].x) + 0.5f * (L1.x - L4x) + sdf * (L2.x - L3x);
        fd[j].y = fmaf(t, aw1[j].y, aw0[j].y) + 0.5f * (L1.y - L4y) + sdf * (L2.y - L3y);
      }

      // G3: [F;dF] @ [Us|V] -> [F@U, F@V; dF@U, junk-for-(1,1)]
      v8f c3 = {};
#pragma unroll
      for (int j = 0; j < 2; ++j) {
        const int k0 = j * 4 + kb;
        v2f bb;
        bb.x = W[uvb[b] + colp + k0 * 8];
        bb.y = W[uvb[b] + colp + (k0 + 1) * 8];
        c3 = wmma4(fd[j], bb, c3);
      }
      // ku = block(0,0) (lanes 0..7);  logj partial = sum V .* (dF@U + F@V)
      float s = 0.f;
#pragma unroll
      for (int r = 0; r < 8; ++r) {
        kout[b][r] = c3[r];
        s = fmaf(c3[r], Vw[b][r], s);
      }
      ss[b] = s;
    }
  };

  auto storeStage = [&](const float (&k)[2][8], float cdt) {
    if (lane < 8) {
#pragma unroll
      for (int r = 0; r < 8; ++r) {
        W[  0 + r * 8 + lane] = fmaf(cdt, k[0][r], Ur[0][r]);
        W[128 + r * 8 + lane] = fmaf(cdt, k[1][r], Ur[1][r]);
      }
    }
    LDS_FENCE();
  };

  auto storeU = [&]() {
    if (lane < 8) {
#pragma unroll
      for (int r = 0; r < 8; ++r) {
        W[  0 + r * 8 + lane] = Ur[0][r];
        W[128 + r * 8 + lane] = Ur[1][r];
      }
    }
    LDS_FENCE();
  };

  // ---- RK4 time loop ----
  const float dt = 1.0f / 16.0f;
  float lj0 = 0.f, lj1 = 0.f;

  for (int step = 0; step < 16; ++step) {
    const float t0 = (float)step * dt;
    float acc[2][8], kt[2][8], ss[2];

    evalRHS(t0, acc, ss);  // k1
    float la0 = ss[0], la1 = ss[1];
    storeStage(acc, 0.5f * dt);

    evalRHS(t0 + 0.5f * dt, kt, ss);  // k2
    la0 += 2.f * ss[0]; la1 += 2.f * ss[1];
#pragma unroll
    for (int b = 0; b < 2; ++b)
#pragma unroll
      for (int r = 0; r < 8; ++r) acc[b][r] += 2.f * kt[b][r];
    storeStage(kt, 0.5f * dt);

    evalRHS(t0 + 0.5f * dt, kt, ss);  // k3
    la0 += 2.f * ss[0]; la1 += 2.f * ss[1];
#pragma unroll
    for (int b = 0; b < 2; ++b)
#pragma unroll
      for (int r = 0; r < 8; ++r) acc[b][r] += 2.f * kt[b][r];
    storeStage(kt, dt);

    evalRHS(t0 + dt, kt, ss);  // k4
    la0 += ss[0]; la1 += ss[1];
#pragma unroll
    for (int b = 0; b < 2; ++b)
#pragma unroll
      for (int r = 0; r < 8; ++r) acc[b][r] += kt[b][r];

    const float w6 = dt / 6.f;
#pragma unroll
    for (int b = 0; b < 2; ++b)
#pragma unroll
      for (int r = 0; r < 8; ++r) Ur[b][r] = fmaf(w6, acc[b][r], Ur[b][r]);
    lj0 = fmaf(w6, la0, lj0);
    lj1 = fmaf(w6, la1, lj1);
    storeU();  // UV holds U for next step's stage 1
  }

  // ---- outputs ----
  if (lane < 8) {
#pragma unroll
    for (int r = 0; r < 8; ++r) {
      outU[(size_t)b0 * 64 + r * 8 + lane] = Ur[0][r];
      outU[(size_t)b1 * 64 + r * 8 + lane] = Ur[1][r];
    }
  }
#pragma unroll
  for (int off = 16; off > 0; off >>= 1) {
    lj0 += __shfl_xor(lj0, off, 32);
    lj1 += __shfl_xor(lj1, off, 32);
  }
  if (lane == 0) {
    outLJ[b0] = lj0;
    outLJ[b1] = lj1;
  }
}

extern "C" void kernel_launch(void* const* d_in, const int* in_sizes, int n_in,
                              void* d_out, int out_size, void* d_ws,
                              size_t ws_size, hipStream_t stream) {
  (void)n_in; (void)out_size; (void)d_ws; (void)ws_size;
  const float* U0  = (const float*)d_in[0];
  const float* eps = (const float*)d_in[1];  // S=1 -> flat [B,8,8]
  const float* W0  = (const float*)d_in[2];
  const float* W1  = (const float*)d_in[3];
  const float* W2  = (const float*)d_in[4];
  const int B = in_sizes[0] / 64;            // 131072

  float* outU  = (float*)d_out;
  float* outLJ = outU + (size_t)B * 64;

  // 2 batch elements per wave, 8 waves (256 threads) per block
  dim3 block(256);
  dim3 grid(B / 16);
  hipLaunchKernelGGL(ode_kernel, grid, block, 0, stream, U0, eps, W0, W1, W2,
                     outU, outLJ);
}